// ModelSGLang_68186900792055
// MI455X (gfx1250) — compile-verified
//
#include <hip/hip_runtime.h>
#include <hip/hip_bf16.h>

// Chunk-local inclusive cumsum along T (chunk = 64) on [B=16, T=16384, H=64] f32.
// Per 16-row block: cumsum = Lt(16x16 lower-tri ones) @ G via
// V_WMMA_F32_16X16X4_F32 (full f32 precision); blocks stitched with a running
// per-column carry broadcast via ds_bpermute_b32. One wave handles a 64x16 slab.
// Streaming data -> non-temporal loads/stores to avoid L2 churn.

typedef __attribute__((ext_vector_type(2))) float v2f;
typedef __attribute__((ext_vector_type(8))) float v8f;

#define B_DIM 16
#define T_DIM 16384
#define H_DIM 64
#define CHUNK 64
#define NCH   (T_DIM / CHUNK)     // 256
#define JT    (H_DIM / 16)        // 4 h-tiles per chunk
#define NWAVES (B_DIM * NCH * JT) // 16384 waves, 1 slab each

__global__ __launch_bounds__(256) void cumsum_chunk_wmma(const float* __restrict__ g,
                                                         float* __restrict__ out) {
  const int wave = (int)((blockIdx.x * blockDim.x + threadIdx.x) >> 5);
  const int lane = (int)(threadIdx.x & 31);
  const int col  = lane & 15;   // h within tile / matrix column N / A-matrix row M
  const int half = lane >> 4;   // lane half selects K offset (A/B) and M offset (C/D)

  // slab -> (b, chunk, h-tile)
  const int j = wave & (JT - 1);
  const int c = (wave >> 2) & (NCH - 1);
  const int b = wave >> 10;     // wave / (JT*NCH)

  const size_t base = ((size_t)b * T_DIM + (size_t)c * CHUNK) * H_DIM
                    + (size_t)(j * 16 + col);
  const float* __restrict__ gp = g + base;
  float* __restrict__ op = out + base;

  // A operands: Lt[m][k] = (k <= m), split into 4 K-subblocks of width 4.
  // A 16x4 f32 layout: lane L holds M = L&15; VGPR v holds K = 2*half + v.
  v2f A[4];
#pragma unroll
  for (int s = 0; s < 4; ++s) {
    A[s].x = (4 * s + 2 * half + 0 <= col) ? 1.0f : 0.0f;
    A[s].y = (4 * s + 2 * half + 1 <= col) ? 1.0f : 0.0f;
  }

  float carry = 0.0f;  // per-column running sum of previous 16-row blocks

#pragma unroll
  for (int I = 0; I < 4; ++I) {
    // B operands: G rows 16I+4s .. +3, cols of this h-tile.
    // B 4x16 f32 layout: lane L holds N = L&15; VGPR v holds K = 2*half + v.
    v2f Bv[4];
#pragma unroll
    for (int s = 0; s < 4; ++s) {
      const int t = 16 * I + 4 * s + 2 * half;
      Bv[s].x = __builtin_nontemporal_load(&gp[(size_t)t * H_DIM]);
      Bv[s].y = __builtin_nontemporal_load(&gp[(size_t)(t + 1) * H_DIM]);
    }

    // D = Lt @ G_I  (local inclusive cumsum of this 16-row block)
    v8f D = {};
#pragma unroll
    for (int s = 0; s < 4; ++s) {
      D = __builtin_amdgcn_wmma_f32_16x16x4_f32(
          /*neg_a=*/false, A[s], /*neg_b=*/false, Bv[s],
          /*c_mod=*/(short)0, D, /*reuse_a=*/false, /*reuse_b=*/false);
    }

    // Add carry from previous blocks (lane == column, uniform across its rows).
    // Block 0 has carry == 0: skip the adds entirely (loop is fully unrolled).
    if (I > 0) {
#pragma unroll
      for (int v = 0; v < 8; ++v) D[v] += carry;
    }

    // New carry = row 15 of this block's output = VGPR7 on lanes 16..31.
    // Broadcast lane (col+16)'s D[7] to both halves. (Dead for I==3 -> DCE'd.)
    if (I < 3) {
      const int cbits = __builtin_amdgcn_ds_bpermute((col + 16) << 2,
                                                     __float_as_int(D[7]));
      carry = __int_as_float(cbits);
    }

    // Store: C/D 16x16 f32 layout: VGPR v -> row v + 8*half, lane -> column.
#pragma unroll
    for (int v = 0; v < 8; ++v) {
      const int t = 16 * I + v + 8 * half;
      __builtin_nontemporal_store(D[v], &op[(size_t)t * H_DIM]);
    }
  }
}

extern "C" void kernel_launch(void* const* d_in, const int* in_sizes, int n_in,
                              void* d_out, int out_size, void* d_ws, size_t ws_size,
                              hipStream_t stream) {
  (void)in_sizes; (void)n_in; (void)d_ws; (void)ws_size; (void)out_size;
  const float* g = (const float*)d_in[0];
  float* out = (float*)d_out;

  const int threads = 256;                       // 8 waves per block
  const int blocks  = (NWAVES * 32) / threads;   // 2048 blocks
  cumsum_chunk_wmma<<<blocks, threads, 0, stream>>>(g, out);
}